// RWKV_26749056320389
// MI455X (gfx1250) — compile-verified
//
#include <hip/hip_runtime.h>
#include <hip/hip_bf16.h>
#include <math.h>

// RWKV single-token forward for MI455X (gfx1250, wave32).
// Bandwidth/latency-bound mat-vec streaming with V_WMMA_F32_16X16X4_F32,
// depth-2 software-pipelined b128 weight streams, batched independent matvecs,
// async-to-LDS staging of the x-vector.

#define RW_L 12
#define RW_E 1024
#define RW_H 4096
#define RW_V 50277

typedef __attribute__((ext_vector_type(2))) float v2f;
typedef __attribute__((ext_vector_type(8))) float v8f;
typedef int v4i __attribute__((vector_size(16)));
typedef __attribute__((address_space(1))) v4i* gbl_v4i_p;
typedef __attribute__((address_space(3))) v4i* lds_v4i_p;

// ---------------------------------------------------------------------------
// init: x = preprocess[token]
// ---------------------------------------------------------------------------
__global__ void rwkv_init_kernel(const float* __restrict__ pre,
                                 const int* __restrict__ tok,
                                 float* __restrict__ x) {
    int i = blockIdx.x * blockDim.x + threadIdx.x;
    if (i < RW_E) x[i] = pre[(size_t)tok[0] * RW_E + i];
}

// ---------------------------------------------------------------------------
// layernorm + token-shift mixes (single block, 256 threads, E=1024)
// ---------------------------------------------------------------------------
__global__ void rwkv_ln_mix_kernel(const float* __restrict__ in,
                                   const float* __restrict__ w,
                                   const float* __restrict__ b,
                                   const float* __restrict__ st,
                                   const float* __restrict__ m0,
                                   const float* __restrict__ m1,
                                   const float* __restrict__ m2,
                                   float* __restrict__ lnout,
                                   float* __restrict__ copy_out,
                                   float* __restrict__ o0,
                                   float* __restrict__ o1,
                                   float* __restrict__ o2) {
    __shared__ float red[256];
    __shared__ float s_mean, s_rstd;
    const int t = threadIdx.x;

    float v0 = in[t + 0];
    float v1 = in[t + 256];
    float v2 = in[t + 512];
    float v3 = in[t + 768];

    red[t] = v0 + v1 + v2 + v3;
    __syncthreads();
    for (int off = 128; off > 0; off >>= 1) {
        if (t < off) red[t] += red[t + off];
        __syncthreads();
    }
    if (t == 0) s_mean = red[0] * (1.0f / (float)RW_E);
    __syncthreads();
    const float m = s_mean;

    float c0 = v0 - m, c1 = v1 - m, c2 = v2 - m, c3 = v3 - m;
    red[t] = c0 * c0 + c1 * c1 + c2 * c2 + c3 * c3;
    __syncthreads();
    for (int off = 128; off > 0; off >>= 1) {
        if (t < off) red[t] += red[t + off];
        __syncthreads();
    }
    if (t == 0)
        s_rstd = rsqrtf(red[0] * (1.0f / (float)RW_E) + 9.999999747378752e-06f);
    __syncthreads();
    const float rs = s_rstd;

#pragma unroll
    for (int j = 0; j < 4; ++j) {
        int i = t + j * 256;
        float xc = in[i] - m;
        float y = xc * rs * w[i] + b[i];
        lnout[i] = y;
        if (copy_out) copy_out[i] = y;
        if (o0) o0[i] = y + m0[i] * st[i];
        if (o1) o1[i] = y + m1[i] * st[i];
        if (o2) o2[i] = y + m2[i] * st[i];
    }
}

// ---------------------------------------------------------------------------
// WKV elementwise: u = w/(d+1e-3), plus bbb/ccc state outputs
// ---------------------------------------------------------------------------
__global__ void rwkv_wkv_kernel(const float* __restrict__ k,
                                const float* __restrict__ v,
                                const float* __restrict__ r,
                                const float* __restrict__ sb,
                                const float* __restrict__ sc,
                                const float* __restrict__ tf,
                                const float* __restrict__ td,
                                float* __restrict__ u,
                                float* __restrict__ bbb,
                                float* __restrict__ ccc) {
    int i = blockIdx.x * blockDim.x + threadIdx.x;
    if (i >= RW_E) return;
    float kk = k[i], vv = v[i], rr = r[i];
    float etf = expf(tf[i]);
    float w = sb[i] + etf * kk * vv;
    float d = sc[i] * rr + etf * kk * rr;
    u[i] = w / (d + 0.001f);
    float etd = expf(td[i]);
    bbb[i] = sb[i] * etd + kk * vv;
    ccc[i] = sc[i] * etd + kk;
}

// ---------------------------------------------------------------------------
// Batched WMMA mat-vec: up to 3 independent jobs in one grid.
// modes: 0 plain | 1 exp | 2 exp+1 | 3 relu^2 | 4 base+v | 5 base + v/dv
// ---------------------------------------------------------------------------
struct MvJob {
    const float* W;
    const float* x;
    float*       y;
    const float* base;
    const float* dv;
    int rows, cols, mode, blk0;
};
struct MvBatch {
    MvJob j[3];
    int   njobs;
};

__device__ __forceinline__ void stage_vec_lds(float* s_x, const float* xvec,
                                              int cols) {
#if __has_builtin(__builtin_amdgcn_global_load_async_to_lds_b128)
    {
        gbl_v4i_p gsrc = (gbl_v4i_p)xvec;  // C-style cast: drops const, AS1
        lds_v4i_p ldst = (lds_v4i_p)s_x;   // generic -> AS3
        const int nchunks = cols >> 2;     // 16B chunks
        for (int i = threadIdx.x; i < nchunks; i += blockDim.x)
            __builtin_amdgcn_global_load_async_to_lds_b128(gsrc + i, ldst + i,
                                                           0, 0);
#if __has_builtin(__builtin_amdgcn_s_wait_asynccnt)
        __builtin_amdgcn_s_wait_asynccnt(0);
#else
        asm volatile("s_wait_asynccnt 0" ::: "memory");
#endif
    }
#else
    for (int i = threadIdx.x; i < cols; i += blockDim.x) s_x[i] = xvec[i];
#endif
    __syncthreads();
}

__global__ void rwkv_matvec_batch_kernel(MvBatch bt) {
    extern __shared__ float s_x[];

    // pick this block's job (uniform per block)
    int jb = 0;
    for (int i = 1; i < bt.njobs; ++i)
        if ((int)blockIdx.x >= bt.j[i].blk0) jb = i;
    const MvJob job  = bt.j[jb];
    const int   lblk = (int)blockIdx.x - job.blk0;
    const int   rows = job.rows;
    const int   cols = job.cols;

    stage_vec_lds(s_x, job.x, cols);

    const int lane = threadIdx.x & 31;
    const int wave = threadIdx.x >> 5;
    const int half = lane >> 4;  // 0: lanes 0-15, 1: lanes 16-31
    const int r16  = lane & 15;

    const int rowBase = (lblk * (int)(blockDim.x >> 5) + wave) * 16;
    int row = rowBase + r16;
    if (row > rows - 1) row = rows - 1;  // clamp ragged tail (stores guarded)

    const float4* wr  = (const float4*)(job.W + (size_t)row * (size_t)cols);
    const float4* sx4 = (const float4*)s_x;

    v8f c = {};
    const int nq = cols >> 3;  // 8 columns per step; nq >= 128 here

    // depth-2 software pipeline
    float4 A0 = wr[half],     B0 = sx4[half];
    float4 A1 = wr[2 + half], B1 = sx4[2 + half];
    for (int q = 0; q < nq - 2; ++q) {
        float4 An = wr[2 * (q + 2) + half];
        float4 Bn = sx4[2 * (q + 2) + half];
        __builtin_prefetch(wr + 2 * q + half + 128, 0, 0);  // ~2KB ahead
        v2f a0 = {A0.x, A0.y}, b0 = {B0.x, B0.y};
        c = __builtin_amdgcn_wmma_f32_16x16x4_f32(false, a0, false, b0,
                                                  (short)0, c, false, false);
        v2f a1 = {A0.z, A0.w}, b1 = {B0.z, B0.w};
        c = __builtin_amdgcn_wmma_f32_16x16x4_f32(false, a1, false, b1,
                                                  (short)0, c, false, false);
        A0 = A1; B0 = B1; A1 = An; B1 = Bn;
    }
    {   // two peeled tail steps (no loads)
        v2f a0 = {A0.x, A0.y}, b0 = {B0.x, B0.y};
        c = __builtin_amdgcn_wmma_f32_16x16x4_f32(false, a0, false, b0,
                                                  (short)0, c, false, false);
        v2f a1 = {A0.z, A0.w}, b1 = {B0.z, B0.w};
        c = __builtin_amdgcn_wmma_f32_16x16x4_f32(false, a1, false, b1,
                                                  (short)0, c, false, false);
        v2f a2 = {A1.x, A1.y}, b2 = {B1.x, B1.y};
        c = __builtin_amdgcn_wmma_f32_16x16x4_f32(false, a2, false, b2,
                                                  (short)0, c, false, false);
        v2f a3 = {A1.z, A1.w}, b3 = {B1.z, B1.w};
        c = __builtin_amdgcn_wmma_f32_16x16x4_f32(false, a3, false, b3,
                                                  (short)0, c, false, false);
    }

    if (r16 == 0) {
        const int rb = rowBase + half * 8;
#pragma unroll
        for (int jj = 0; jj < 8; ++jj) {
            int rr = rb + jj;
            if (rr < rows) {
                float v = c[jj];
                float o;
                switch (job.mode) {
                    case 1: o = expf(v); break;
                    case 2: o = expf(v) + 1.0f; break;
                    case 3: { float tt = v > 0.0f ? v : 0.0f; o = tt * tt; } break;
                    case 4: o = job.base[rr] + v; break;
                    case 5: o = job.base[rr] + v / job.dv[rr]; break;
                    default: o = v; break;
                }
                job.y[rr] = o;
            }
        }
    }
}

// ---------------------------------------------------------------------------
// host side
// ---------------------------------------------------------------------------
static void launch_batch(const MvJob* jobs, int n, hipStream_t stream) {
    MvBatch bt{};
    int blocks = 0, maxc = 0;
    for (int i = 0; i < n; ++i) {
        bt.j[i] = jobs[i];
        bt.j[i].blk0 = blocks;
        blocks += (jobs[i].rows + 63) / 64;  // 128 thr = 4 waves = 64 rows
        if (jobs[i].cols > maxc) maxc = jobs[i].cols;
    }
    bt.njobs = n;
    rwkv_matvec_batch_kernel<<<dim3(blocks), dim3(128),
                               maxc * sizeof(float), stream>>>(bt);
}

extern "C" void kernel_launch(void* const* d_in, const int* in_sizes, int n_in,
                              void* d_out, int out_size, void* d_ws, size_t ws_size,
                              hipStream_t stream) {
    const float* statea     = (const float*)d_in[0];
    const float* stateb     = (const float*)d_in[1];
    const float* statec     = (const float*)d_in[2];
    const float* stated     = (const float*)d_in[3];
    const float* preprocess = (const float*)d_in[4];
    const float* ln1w       = (const float*)d_in[5];
    const float* ln1b       = (const float*)d_in[6];
    const float* ln2w       = (const float*)d_in[7];
    const float* ln2b       = (const float*)d_in[8];
    const float* att_key    = (const float*)d_in[9];
    const float* att_rec    = (const float*)d_in[10];
    const float* att_val    = (const float*)d_in[11];
    const float* kktk       = (const float*)d_in[12];
    const float* vvtv       = (const float*)d_in[13];
    const float* rrtr       = (const float*)d_in[14];
    const float* time_first = (const float*)d_in[15];
    const float* time_decay = (const float*)d_in[16];
    const float* outputvv   = (const float*)d_in[17];
    const float* tmk        = (const float*)d_in[18];
    const float* tmr        = (const float*)d_in[19];
    const float* key_ffn    = (const float*)d_in[20];
    const float* rec_ffn    = (const float*)d_in[21];
    const float* val_ffn    = (const float*)d_in[22];
    const float* post0      = (const float*)d_in[23];
    const float* post1      = (const float*)d_in[24];
    const float* post2      = (const float*)d_in[25];
    const int*   token      = (const int*)d_in[26];

    float* out    = (float*)d_out;
    float* logits = out;               // [V]
    float* aaa    = out + RW_V;        // [L,E]
    float* bbb    = aaa + RW_L * RW_E; // [L,E]
    float* ccc    = bbb + RW_L * RW_E; // [L,E]
    float* ddd    = ccc + RW_L * RW_E; // [L,E]

    float* ws  = (float*)d_ws;
    float* x   = ws;              // [E] hidden state
    float* xy  = x   + RW_E;      // [E] LN1 output
    float* xk  = xy  + RW_E;      // [E]
    float* xv  = xk  + RW_E;      // [E]
    float* xr  = xv  + RW_E;      // [E]
    float* kv  = xr  + RW_E;      // [E] k
    float* vv_ = kv  + RW_E;      // [E] v
    float* rv  = vv_ + RW_E;      // [E] r
    float* uu  = rv  + RW_E;      // [E] w/(d+eps)
    float* sxx = uu  + RW_E;      // [E]
    float* xx  = sxx + RW_E;      // [E] LN2 output
    float* xkf = xx  + RW_E;      // [E]
    float* xrf = xkf + RW_E;      // [E]
    float* km  = xrf + RW_E;      // [H]
    float* rt  = km  + RW_H;      // [E]
    float* xf  = rt  + RW_E;      // [E] final LN

    rwkv_init_kernel<<<dim3((RW_E + 255) / 256), dim3(256), 0, stream>>>(
        preprocess, token, x);

    for (int l = 0; l < RW_L; ++l) {
        const size_t oE  = (size_t)l * RW_E;
        const size_t oEE = (size_t)l * RW_E * RW_E;
        const size_t oHE = (size_t)l * RW_H * RW_E;
        const size_t oEH = (size_t)l * RW_E * RW_H;

        // xy = LN1(x); aaa = xy; xk/xv/xr = xy + mix*sa
        rwkv_ln_mix_kernel<<<dim3(1), dim3(256), 0, stream>>>(
            x, ln1w + oE, ln1b + oE, statea + oE,
            kktk + oE, vvtv + oE, rrtr + oE,
            xy, aaa + oE, xk, xv, xr);

        // k = exp(ak@xk); v = av@xv; r = exp(ar@xr)+1  (one fused grid)
        {
            MvJob jobs[3] = {
                {att_key + oEE, xk, kv,  nullptr, nullptr, RW_E, RW_E, 1, 0},
                {att_val + oEE, xv, vv_, nullptr, nullptr, RW_E, RW_E, 0, 0},
                {att_rec + oEE, xr, rv,  nullptr, nullptr, RW_E, RW_E, 2, 0},
            };
            launch_batch(jobs, 3, stream);
        }

        rwkv_wkv_kernel<<<dim3(RW_E / 256), dim3(256), 0, stream>>>(
            kv, vv_, rv, stateb + oE, statec + oE,
            time_first + oE, time_decay + oE,
            uu, bbb + oE, ccc + oE);

        // sxx = x + outputvv @ u
        {
            MvJob jobs[1] = {
                {outputvv + oEE, uu, sxx, x, nullptr, RW_E, RW_E, 4, 0},
            };
            launch_batch(jobs, 1, stream);
        }

        // xx = LN2(sxx); ddd = xx; xkf/xrf = xx + mix*sd
        rwkv_ln_mix_kernel<<<dim3(1), dim3(256), 0, stream>>>(
            sxx, ln2w + oE, ln2b + oE, stated + oE,
            tmk + oE, tmr + oE, nullptr,
            xx, ddd + oE, xkf, xrf, nullptr);

        // km = relu(kf@xkf)^2; rt = exp(rf@xrf)+1  (one fused grid)
        {
            MvJob jobs[2] = {
                {key_ffn + oHE, xkf, km, nullptr, nullptr, RW_H, RW_E, 3, 0},
                {rec_ffn + oEE, xrf, rt, nullptr, nullptr, RW_E, RW_E, 2, 0},
            };
            launch_batch(jobs, 2, stream);
        }

        // x = sxx + (val_ffn @ km) / rt
        {
            MvJob jobs[1] = {
                {val_ffn + oEH, km, x, sxx, rt, RW_E, RW_H, 5, 0},
            };
            launch_batch(jobs, 1, stream);
        }
    }

    // xf = LN(x, post0, post1); logits = post2 @ xf
    rwkv_ln_mix_kernel<<<dim3(1), dim3(256), 0, stream>>>(
        x, post0, post1, nullptr, nullptr, nullptr, nullptr,
        xf, nullptr, nullptr, nullptr, nullptr);
    {
        MvJob jobs[1] = {
            {post2, xf, logits, nullptr, nullptr, RW_V, RW_E, 0, 0},
        };
        launch_batch(jobs, 1, stream);
    }
}